// PScanTriton_24412594111197
// MI455X (gfx1250) — compile-verified
//
#include <hip/hip_runtime.h>
#include <hip/hip_bf16.h>

typedef __attribute__((ext_vector_type(2))) float v2f;
typedef __attribute__((ext_vector_type(8))) float v8f;

#define BB 4
#define LL 2048
#define CC 64
#define RR 8
#define NC 16              // chunks along L
#define TT (LL / NC)       // 128 steps per chunk
#define NCHAN (BB * CC)    // 256 channels
#define NAGG (NCHAN * NC)  // 4096 (channel, chunk) waves

// Wave-uniform broadcast from a fixed lane: lowers to v_readlane_b32 (SGPR
// result) instead of ds_bpermute — keeps the 16-wide state broadcast off the
// LDS pipeline in the streaming loops.
__device__ __forceinline__ float bcast_lane(float v, int srclane) {
  return __int_as_float(__builtin_amdgcn_readlane(__float_as_int(v), srclane));
}

// ---------------------------------------------------------------------------
// Pass 1: per (channel, chunk) wave, compute
//   P_c = M_{t1} ... M_{t0}   (16x16 real embedding of the complex 8x8 chain)
//   v_c = chunk-local end state with zero initial state
// Running product lives in LDS; each step = 4x v_wmma_f32_16x16x4_f32.
// Single-wave workgroup => __syncthreads() is an S_NOP barrier (compiler fence).
// ---------------------------------------------------------------------------
__global__ __launch_bounds__(32) void pscan_chunk_reduce(
    const float* __restrict__ Ar, const float* __restrict__ Ai,
    const float* __restrict__ Xr, const float* __restrict__ Xi,
    float* __restrict__ wsP, float* __restrict__ wsV) {
  __shared__ float P[256];  // 16x16 running product, row-major

  const int lane = threadIdx.x & 31;
  const int wave = blockIdx.x;  // one wave per block
  const int chunk = wave % NC;
  const int chan = wave / NC;
  const int b = chan / CC, c = chan % CC;
  const int m = lane & 15;   // output row (A operand M)
  const int h = lane >> 4;   // lane half (selects K pair 2h,2h+1)
  const int mm = m & 7;
  const bool low = (m < 8);

  // identity product
  for (int i = lane; i < 256; i += 32) P[i] = (i % 17 == 0) ? 1.0f : 0.0f;

  // Per-lane source descriptors for the 4 K-slices of the embedding
  //   M = [[Ar, -Ai], [Ai, Ar]]
  // A-operand layout: lane = m + 16*(K>=2 in slice), element k = 4*sc + 2h + vg
  int off[4];
  bool fromRe[4];
  float sgn[4];
#pragma unroll
  for (int sc = 0; sc < 4; ++sc) {
    const int k0 = 4 * sc + 2 * h;
    if (sc < 2) {  // k < 8 region
      off[sc] = mm * 8 + k0;
      fromRe[sc] = low;          // low rows: Ar ; high rows: Ai
      sgn[sc] = 1.0f;
    } else {       // k >= 8 region
      off[sc] = mm * 8 + (k0 - 8);
      fromRe[sc] = !low;         // low rows: -Ai ; high rows: Ar
      sgn[sc] = low ? -1.0f : 1.0f;
    }
  }

  float y[16];
#pragma unroll
  for (int k = 0; k < 16; ++k) y[k] = 0.0f;
  float ymine = 0.0f;

  size_t baseA = ((size_t)(b * LL + chunk * TT) * CC + c) * 64;
  size_t baseX = ((size_t)(b * LL + chunk * TT) * CC + c) * 8;
  const size_t strA = (size_t)CC * 64, strX = (size_t)CC * 8;

  for (int t = 0; t < TT; ++t) {
    const float* pr = Ar + baseA;
    const float* pi = Ai + baseA;
    __builtin_prefetch(pr + strA + lane * 2, 0, 1);
    __builtin_prefetch(pi + strA + lane * 2, 0, 1);

    // Build the 4 A-operand slices of M_t (signs folded in) and, from the same
    // registers, the per-lane partial of the matvec  y' = M_t y + x_t.
    v2f a[4];
    float part = 0.0f;
#pragma unroll
    for (int sc = 0; sc < 4; ++sc) {
      const float* src = (fromRe[sc] ? pr : pi) + off[sc];
      float2 v = *(const float2*)src;
      const float ax = v.x * sgn[sc];
      const float ay = v.y * sgn[sc];
      a[sc].x = ax;
      a[sc].y = ay;
      const float ya = h ? y[4 * sc + 2] : y[4 * sc + 0];
      const float yb = h ? y[4 * sc + 3] : y[4 * sc + 1];
      part = fmaf(ax, ya, fmaf(ay, yb, part));
    }
    const float xm = low ? Xr[baseX + m] : Xi[baseX + mm];

    __syncthreads();  // previous step's P writeback visible to all lanes
    // B operand: rows 4sc..4sc+3 of P; layout K = 2h+vg, N = lane%16
    v2f bv[4];
#pragma unroll
    for (int sc = 0; sc < 4; ++sc) {
      const int kb = 4 * sc + 2 * h;
      bv[sc].x = P[kb * 16 + m];
      bv[sc].y = P[(kb + 1) * 16 + m];
    }
    __syncthreads();  // all B reads done before overwriting P (WAR)

    v8f acc = {0.f, 0.f, 0.f, 0.f, 0.f, 0.f, 0.f, 0.f};
#pragma unroll
    for (int sc = 0; sc < 4; ++sc)
      acc = __builtin_amdgcn_wmma_f32_16x16x4_f32(false, a[sc], false, bv[sc],
                                                  (short)0, acc, false, false);

    // D layout: VGPR r holds row (r + 8h), col = lane%16  ->  row-major LDS
#pragma unroll
    for (int r = 0; r < 8; ++r) P[(r + 8 * h) * 16 + m] = acc[r];

    // finish the state update: combine halves, add x, broadcast via readlane
    const float tot = part + __shfl_xor(part, 16, 32);
    const float yn = tot + xm;
#pragma unroll
    for (int k = 0; k < 16; ++k) y[k] = bcast_lane(yn, k);
    ymine = yn;

    baseA += strA;
    baseX += strX;
  }

  __syncthreads();
  float* dP = wsP + (size_t)wave * 256;
  for (int i = lane; i < 256; i += 32) dP[i] = P[i];
  if (h == 0) wsV[(size_t)wave * 16 + m] = ymine;
}

// ---------------------------------------------------------------------------
// Pass 2: per-channel sequential combine over the NC chunk aggregates:
//   s_c = P_c s_{c-1} + v_c ; emit carry-in (= s_{c-1}) for each chunk.
// ---------------------------------------------------------------------------
__global__ __launch_bounds__(32) void pscan_carry_scan(
    const float* __restrict__ wsP, const float* __restrict__ wsV,
    float* __restrict__ wsS) {
  const int lane = threadIdx.x & 31;
  const int chan = blockIdx.x;
  const int m = lane & 15, h = lane >> 4;

  float s[16];
#pragma unroll
  for (int k = 0; k < 16; ++k) s[k] = 0.0f;
  float smine = 0.0f;

  for (int cI = 0; cI < NC; ++cI) {
    const size_t agg = (size_t)chan * NC + cI;
    if (h == 0) wsS[agg * 16 + m] = smine;  // carry-in for this chunk
    const float* prow = wsP + agg * 256 + m * 16 + 8 * h;
    const float4 p0 = *(const float4*)prow;
    const float4 p1 = *(const float4*)(prow + 4);
    float part = 0.f;
    part = fmaf(p0.x, h ? s[8] : s[0], part);
    part = fmaf(p0.y, h ? s[9] : s[1], part);
    part = fmaf(p0.z, h ? s[10] : s[2], part);
    part = fmaf(p0.w, h ? s[11] : s[3], part);
    part = fmaf(p1.x, h ? s[12] : s[4], part);
    part = fmaf(p1.y, h ? s[13] : s[5], part);
    part = fmaf(p1.z, h ? s[14] : s[6], part);
    part = fmaf(p1.w, h ? s[15] : s[7], part);
    const float tot = part + __shfl_xor(part, 16, 32);
    const float sn = tot + wsV[agg * 16 + m];
#pragma unroll
    for (int k = 0; k < 16; ++k) s[k] = bcast_lane(sn, k);
    smine = sn;
  }
}

// ---------------------------------------------------------------------------
// Pass 3: replay each chunk from its carry-in, streaming A once more with
// b128 row loads; write interleaved (re,im) outputs [B,L,C,R,2] as b64 pairs.
// ---------------------------------------------------------------------------
__global__ __launch_bounds__(256) void pscan_emit(
    const float* __restrict__ Ar, const float* __restrict__ Ai,
    const float* __restrict__ Xr, const float* __restrict__ Xi,
    const float* __restrict__ wsS, float* __restrict__ out) {
  const int tid = blockIdx.x * blockDim.x + threadIdx.x;
  const int wave = tid >> 5;
  const int lane = threadIdx.x & 31;
  const int chunk = wave % NC;
  const int chan = wave / NC;
  const int b = chan / CC, c = chan % CC;
  const int m = lane & 15, h = lane >> 4, mm = m & 7;
  const bool low = (m < 8);
  const float sgn2 = low ? -1.0f : 1.0f;  // second half of row is -Ai for low rows

  const float mine0 = wsS[((size_t)chan * NC + chunk) * 16 + m];
  float y[16];
#pragma unroll
  for (int k = 0; k < 16; ++k) y[k] = bcast_lane(mine0, k);

  size_t baseA = ((size_t)(b * LL + chunk * TT) * CC + c) * 64;
  size_t baseX = ((size_t)(b * LL + chunk * TT) * CC + c) * 8;
  size_t baseO = ((size_t)(b * LL + chunk * TT) * CC + c) * 16;
  const size_t strA = (size_t)CC * 64, strX = (size_t)CC * 8, strO = (size_t)CC * 16;
  const int rowoff = mm * 8;

  for (int t = 0; t < TT; ++t) {
    const float* pr = Ar + baseA;
    const float* pi = Ai + baseA;
    const float* p1 = (low ? pr : pi) + rowoff;  // first 8 cols of row m of M
    const float* p2 = (low ? pi : pr) + rowoff;  // last 8 cols (sign via sgn2)
    __builtin_prefetch(p1 + strA, 0, 1);
    __builtin_prefetch(p2 + strA, 0, 1);
    const float4 r10 = *(const float4*)p1;
    const float4 r11 = *(const float4*)(p1 + 4);
    const float4 r20 = *(const float4*)p2;
    const float4 r21 = *(const float4*)(p2 + 4);

    float a1 = 0.f, a2 = 0.f;
    a1 = fmaf(r10.x, y[0], a1);  a1 = fmaf(r10.y, y[1], a1);
    a1 = fmaf(r10.z, y[2], a1);  a1 = fmaf(r10.w, y[3], a1);
    a1 = fmaf(r11.x, y[4], a1);  a1 = fmaf(r11.y, y[5], a1);
    a1 = fmaf(r11.z, y[6], a1);  a1 = fmaf(r11.w, y[7], a1);
    a2 = fmaf(r20.x, y[8], a2);  a2 = fmaf(r20.y, y[9], a2);
    a2 = fmaf(r20.z, y[10], a2); a2 = fmaf(r20.w, y[11], a2);
    a2 = fmaf(r21.x, y[12], a2); a2 = fmaf(r21.y, y[13], a2);
    a2 = fmaf(r21.z, y[14], a2); a2 = fmaf(r21.w, y[15], a2);

    const float xm = low ? Xr[baseX + m] : Xi[baseX + mm];
    const float yn = fmaf(sgn2, a2, a1) + xm;

    // pair (re_i, im_i) and store as b64: lane i<8 stores {y_i, y_{8+i}}
    const float partner = __shfl_xor(yn, 8, 32);
    if (lane < 8) {
      float2 o2;
      o2.x = yn;
      o2.y = partner;
      *(float2*)(out + baseO + 2 * m) = o2;
    }
#pragma unroll
    for (int k = 0; k < 16; ++k) y[k] = bcast_lane(yn, k);

    baseA += strA;
    baseX += strX;
    baseO += strO;
  }
}

// ---------------------------------------------------------------------------
extern "C" void kernel_launch(void* const* d_in, const int* in_sizes, int n_in,
                              void* d_out, int out_size, void* d_ws, size_t ws_size,
                              hipStream_t stream) {
  (void)in_sizes; (void)n_in; (void)out_size; (void)ws_size;
  const float* Ar = (const float*)d_in[0];
  const float* Ai = (const float*)d_in[1];
  const float* Xr = (const float*)d_in[2];
  const float* Xi = (const float*)d_in[3];
  float* out = (float*)d_out;

  // workspace: P (NAGG*256) | V (NAGG*16) | S (NAGG*16)  => ~4.6 MB of f32
  float* wsP = (float*)d_ws;
  float* wsV = wsP + (size_t)NAGG * 256;
  float* wsS = wsV + (size_t)NAGG * 16;

  pscan_chunk_reduce<<<NAGG, 32, 0, stream>>>(Ar, Ai, Xr, Xi, wsP, wsV);
  pscan_carry_scan<<<NCHAN, 32, 0, stream>>>(wsP, wsV, wsS);
  pscan_emit<<<(NAGG * 32) / 256, 256, 0, stream>>>(Ar, Ai, Xr, Xi, wsS, out);
}